// Encoder_33423435498428
// MI455X (gfx1250) — compile-verified
//
#include <hip/hip_runtime.h>
#include <math.h>

#define HH 192
#define WW 192
#define LSZ (HH*WW)      // 36864
#define BB 4
#define KK 4
#define NC 64            // scan chunks
#define CL (LSZ/NC)      // 576 steps per chunk
#define TT 32            // LDS tile of scan steps
#define SEG 36           // t-segments per (b,k) in WMMA kernel
#define TPW 8            // 16-wide t-tiles per wave in WMMA kernel

typedef float v2f __attribute__((ext_vector_type(2)));
typedef float v8f __attribute__((ext_vector_type(8)));

__device__ __forceinline__ float siluf(float x) { return x / (1.f + __expf(-x)); }
// softplus(x) = max(x,0) + log1p(exp(-|x|)); fast-math form, branchless, fp32-stable
__device__ __forceinline__ float softplusf(float x) {
  return fmaxf(x, 0.f) + __logf(1.f + __expf(-fabsf(x)));
}

// map scan-order index t of direction k to spatial index l = h*W + w
__device__ __forceinline__ int map_l(int k, int t) {
  int tt = (k & 2) ? (LSZ - 1 - t) : t;
  if (k & 1) { int w = tt / HH; int h = tt - w * HH; return h * WW + w; }
  return tt;
}

// async memory -> LDS, one dword per lane (ASYNCcnt-tracked, gfx1250)
__device__ __forceinline__ void async_ld(const float* g, float* lds) {
  unsigned l32 = (unsigned)(unsigned long long)(void*)lds;
  asm volatile("global_load_async_to_lds_b32 %0, %1, off"
               :: "v"(l32), "v"((unsigned long long)g) : "memory");
}

// ---------------- kernel 0: 1x1 input conv (1 -> 8), NCHW -> BHWC ----------------
__global__ void k_in_proj(const float* __restrict__ img, const float* __restrict__ w_in,
                          const float* __restrict__ b_in, float* __restrict__ feat) {
  int idx = blockIdx.x * blockDim.x + threadIdx.x;
  if (idx >= BB * LSZ) return;
  float v = img[idx];
  float* o = feat + idx * 8;
#pragma unroll
  for (int c = 0; c < 8; ++c) o[c] = v * w_in[c] + b_in[c];
}

// ---------------- kernel 1: LN(8) + in_proj (8->32), split into xc (B,16,L) and z (B,L,16) ----
__global__ void k_ln_inproj(const float* __restrict__ feat, const float* __restrict__ lnw,
                            const float* __restrict__ lnb, const float* __restrict__ inw,
                            float* __restrict__ xc, float* __restrict__ z) {
  int idx = blockIdx.x * blockDim.x + threadIdx.x;
  if (idx >= BB * LSZ) return;
  int b = idx / LSZ, l = idx - b * LSZ;
  const float* x = feat + idx * 8;
  float xv[8], m = 0.f;
#pragma unroll
  for (int c = 0; c < 8; ++c) { xv[c] = x[c]; m += xv[c]; }
  m *= 0.125f;
  float var = 0.f;
#pragma unroll
  for (int c = 0; c < 8; ++c) { float d = xv[c] - m; var += d * d; }
  var *= 0.125f;
  float rs = rsqrtf(var + 1e-5f);
  float xn[8];
#pragma unroll
  for (int c = 0; c < 8; ++c) xn[c] = (xv[c] - m) * rs * lnw[c] + lnb[c];
#pragma unroll
  for (int j = 0; j < 32; ++j) {
    float s = 0.f;
#pragma unroll
    for (int c = 0; c < 8; ++c) s += xn[c] * inw[j * 8 + c];
    if (j < 16) xc[(b * 16 + j) * LSZ + l] = s;
    else        z[idx * 16 + (j - 16)] = s;
  }
}

// ---------------- kernel 2: depthwise 3x3 conv (SAME) + bias + SiLU --------------
__global__ void k_dwconv(const float* __restrict__ xc, const float* __restrict__ cw,
                         const float* __restrict__ cb, float* __restrict__ xo) {
  int idx = blockIdx.x * blockDim.x + threadIdx.x;
  if (idx >= BB * 16 * LSZ) return;
  int l = idx % LSZ; int bc = idx / LSZ; int c = bc & 15;
  int h = l / WW, w = l - h * WW;
  const float* in = xc + bc * LSZ;
  const float* k9 = cw + c * 9;
  float s = cb[c];
#pragma unroll
  for (int kh = 0; kh < 3; ++kh) {
    int hh = h + kh - 1;
    if (hh < 0 || hh >= HH) continue;
#pragma unroll
    for (int kw = 0; kw < 3; ++kw) {
      int wp = w + kw - 1;
      if (wp < 0 || wp >= WW) continue;
      s += in[hh * WW + wp] * k9[kh * 3 + kw];
    }
  }
  xo[idx] = siluf(s);
}

// ------- kernel 3: x_dbl projection via V_WMMA_F32_16X16X4_F32, fused delta -------
// per (b,k): out[c,t] = sum_d xpw[k,c,d] * xs_k[d,t]  (c=0..32, d=0..15)
// c==0 -> dts -> delta = softplus(dts*dtw + dtb); c in [1,32] -> BC rows 0..31 (Bm,Cm)
__global__ void __launch_bounds__(256) k_xdbl(
    const float* __restrict__ xconv, const float* __restrict__ xpw,
    const float* __restrict__ dtw, const float* __restrict__ dtb,
    float* __restrict__ BC, float* __restrict__ delta) {
  int bk = blockIdx.x / SEG, seg = blockIdx.x % SEG;
  int b = bk >> 2, k = bk & 3;
  int lane = threadIdx.x & 31;
  int wv = threadIdx.x >> 5;
  int row = lane & 15, half = lane >> 4;

  // A operands (xpw), kept in registers for all t-tiles of this wave.
  // 16x4 f32 A layout: lanes 0-15 rows, VGPR0=K0,VGPR1=K1; lanes 16-31: K2,K3.
  v2f A[3][4];
#pragma unroll
  for (int ct = 0; ct < 3; ++ct)
#pragma unroll
    for (int ks = 0; ks < 4; ++ks) {
      int c = ct * 16 + row;
      int d = ks * 4 + half * 2;
      float a0 = 0.f, a1 = 0.f;
      if (c < 33) { a0 = xpw[(k * 33 + c) * 16 + d]; a1 = xpw[(k * 33 + c) * 16 + d + 1]; }
      v2f t; t.x = a0; t.y = a1; A[ct][ks] = t;
    }

  const float* xcb = xconv + b * 16 * LSZ;
  int t0base = (seg * 8 + wv) * (TPW * 16);
  for (int ti = 0; ti < TPW; ++ti) {
    int t0 = t0base + ti * 16;
    int tcol = t0 + row;
    int l = map_l(k, tcol);
    v8f acc0 = {0.f,0.f,0.f,0.f,0.f,0.f,0.f,0.f};
    v8f acc1 = acc0, acc2 = acc0;
#pragma unroll
    for (int ks = 0; ks < 4; ++ks) {
      // 4x16 f32 B layout: VGPR0: rows K0 (lanes 0-15) / K2 (16-31); VGPR1: K1/K3.
      int dbase = ks * 4 + half * 2;
      v2f Bop; Bop.x = xcb[dbase * LSZ + l]; Bop.y = xcb[(dbase + 1) * LSZ + l];
      acc0 = __builtin_amdgcn_wmma_f32_16x16x4_f32(false, A[0][ks], false, Bop, (short)0, acc0, false, false);
      acc1 = __builtin_amdgcn_wmma_f32_16x16x4_f32(false, A[1][ks], false, Bop, (short)0, acc1, false, false);
      acc2 = __builtin_amdgcn_wmma_f32_16x16x4_f32(false, A[2][ks], false, Bop, (short)0, acc2, false, false);
    }
    // dts row (c==0) lives in acc0 elem 0 on lanes 0-15; broadcast per column.
    float dts = __shfl(acc0[0], row, 32);
#pragma unroll
    for (int j = 0; j < 8; ++j) {
      int cd = half * 8 + j;
      float pre = dts * dtw[k * 16 + cd] + dtb[k * 16 + cd];
      delta[(bk * 16 + cd) * LSZ + tcol] = softplusf(pre);
    }
    // C/D layout: elem j -> matrix row j (lanes 0-15) / j+8 (lanes 16-31), col = lane&15
    float* bcp = BC + (size_t)bk * 32 * LSZ;
#pragma unroll
    for (int j = 0; j < 8; ++j) {
      int c0 = half * 8 + j;
      int c = c0;                   // ctile 0: c = 0..15
      if (c >= 1) bcp[(c - 1) * LSZ + tcol] = acc0[j];
      c = 16 + c0;                  // ctile 1: c = 16..31
      bcp[(c - 1) * LSZ + tcol] = acc1[j];
      c = 32 + c0;                  // ctile 2: only c == 32 is real
      if (c == 32) bcp[31 * LSZ + tcol] = acc2[j];
    }
  }
}

// -------- kernel 4: scan phase 1 — per-chunk (prod a, local h), thread = (c,n) -----
// Double-buffered async (memory->LDS) pipeline: 6 async ops per wave per tile.
__global__ void __launch_bounds__(256) k_scan1(
    const float* __restrict__ delta, const float* __restrict__ xconv,
    const float* __restrict__ BC, const float* __restrict__ alog,
    float* __restrict__ chA, float* __restrict__ chH) {
  int blk = blockIdx.x;              // bk*NC + chunk
  int bk = blk / NC, ch = blk - bk * NC;
  int b = bk >> 2, k = bk & 3;
  int tid = threadIdx.x;
  int c = tid >> 4, n = tid & 15;
  float Ac = -__expf(alog[(k * 16 + c) * 16 + n]);
  const float* drow = delta + (size_t)bk * 16 * LSZ;
  const float* urow = xconv + (size_t)b * 16 * LSZ;
  const float* brow = BC + (size_t)bk * 32 * LSZ;   // Bm = rows 0..15
  __shared__ float sd[2][16][TT + 1], su[2][16][TT + 1], sb[2][16][TT + 1];

  auto issue = [&](int bf, int tb) {
    for (int i = tid; i < 48 * TT; i += 256) {   // 6 per thread (uniform per wave)
      int r = i >> 5, col = i & 31;
      int t = tb + col;
      const float* g; float* lp;
      if (r < 16)      { g = drow + r * LSZ + t;                lp = &sd[bf][r][col]; }
      else if (r < 32) { g = urow + (r - 16) * LSZ + map_l(k, t); lp = &su[bf][r - 16][col]; }
      else             { g = brow + (r - 32) * LSZ + t;          lp = &sb[bf][r - 32][col]; }
      async_ld(g, lp);
    }
  };

  float h = 0.f, Ap = 1.f;
  int tbeg = ch * CL, tend = tbeg + CL;
  int buf = 0;
  issue(0, tbeg);
  for (int tb = tbeg; tb < tend; tb += TT) {
    if (tb + TT < tend) {
      issue(buf ^ 1, tb + TT);
      asm volatile("s_wait_asynccnt 0x6" ::: "memory");  // oldest tile landed in LDS
    } else {
      asm volatile("s_wait_asynccnt 0x0" ::: "memory");
    }
    __syncthreads();
#pragma unroll 4
    for (int col = 0; col < TT; ++col) {
      float d = sd[buf][c][col];
      float a = __expf(d * Ac);
      h = a * h + d * su[buf][c][col] * sb[buf][n][col];
      Ap *= a;
    }
    __syncthreads();
    buf ^= 1;
  }
  chA[blk * 256 + tid] = Ap;
  chH[blk * 256 + tid] = h;
}

// -------- kernel 5: scan phase 2 — cross-chunk prefix, per (b,k) -------------------
__global__ void k_scan2(const float* __restrict__ chA, const float* __restrict__ chH,
                        float* __restrict__ chH0) {
  int bk = blockIdx.x, tid = threadIdx.x;
  float run = 0.f;
  for (int ch = 0; ch < NC; ++ch) {
    int idx = (bk * NC + ch) * 256 + tid;
    chH0[idx] = run;
    run = chA[idx] * run + chH[idx];
  }
}

// -------- kernel 6: scan phase 3 — replay with init state, emit y = sum_n h*C ------
// Double-buffered async pipeline: 8 async ops per wave per tile.
__global__ void __launch_bounds__(256) k_scan3(
    const float* __restrict__ delta, const float* __restrict__ xconv,
    const float* __restrict__ BC, const float* __restrict__ alog,
    const float* __restrict__ chH0, float* __restrict__ out_y) {
  int blk = blockIdx.x;
  int bk = blk / NC, ch = blk - bk * NC;
  int b = bk >> 2, k = bk & 3;
  int tid = threadIdx.x;
  int c = tid >> 4, n = tid & 15;
  float Ac = -__expf(alog[(k * 16 + c) * 16 + n]);
  const float* drow = delta + (size_t)bk * 16 * LSZ;
  const float* urow = xconv + (size_t)b * 16 * LSZ;
  const float* brow = BC + (size_t)bk * 32 * LSZ;
  const float* crow = brow + 16 * LSZ;              // Cm = rows 16..31
  __shared__ float sd[2][16][TT + 1], su[2][16][TT + 1], sb[2][16][TT + 1], sc[2][16][TT + 1];

  auto issue = [&](int bf, int tb) {
    for (int i = tid; i < 64 * TT; i += 256) {   // 8 per thread (uniform per wave)
      int r = i >> 5, col = i & 31;
      int t = tb + col;
      const float* g; float* lp;
      if (r < 16)      { g = drow + r * LSZ + t;                  lp = &sd[bf][r][col]; }
      else if (r < 32) { g = urow + (r - 16) * LSZ + map_l(k, t); lp = &su[bf][r - 16][col]; }
      else if (r < 48) { g = brow + (r - 32) * LSZ + t;           lp = &sb[bf][r - 32][col]; }
      else             { g = crow + (r - 48) * LSZ + t;           lp = &sc[bf][r - 48][col]; }
      async_ld(g, lp);
    }
  };

  float h = chH0[blk * 256 + tid];
  int tbeg = ch * CL, tend = tbeg + CL;
  int buf = 0;
  issue(0, tbeg);
  for (int tb = tbeg; tb < tend; tb += TT) {
    if (tb + TT < tend) {
      issue(buf ^ 1, tb + TT);
      asm volatile("s_wait_asynccnt 0x8" ::: "memory");
    } else {
      asm volatile("s_wait_asynccnt 0x0" ::: "memory");
    }
    __syncthreads();
    for (int col = 0; col < TT; ++col) {
      float d = sd[buf][c][col];
      float a = __expf(d * Ac);
      h = a * h + d * su[buf][c][col] * sb[buf][n][col];
      float p = h * sc[buf][n][col];
      p += __shfl_xor(p, 8, 16);
      p += __shfl_xor(p, 4, 16);
      p += __shfl_xor(p, 2, 16);
      p += __shfl_xor(p, 1, 16);
      if (n == 0) out_y[(size_t)(bk * 16 + c) * LSZ + (tb + col)] = p;
    }
    __syncthreads();
    buf ^= 1;
  }
}

// ---- kernel 7: gather 4 directions + D-term, LN(16), gate by SiLU(z), out_proj, residual
__global__ void k_combine(const float* __restrict__ out_y, const float* __restrict__ xconv,
                          const float* __restrict__ z, const float* __restrict__ ds,
                          const float* __restrict__ onw, const float* __restrict__ onb,
                          const float* __restrict__ opw, const float* __restrict__ featIn,
                          float* __restrict__ featOut) {
  int idx = blockIdx.x * blockDim.x + threadIdx.x;
  if (idx >= BB * LSZ) return;
  int b = idx / LSZ, l = idx - b * LSZ;
  int h = l / WW, w = l - h * WW;
  int t1 = w * HH + h;
  int t0 = l, t2 = LSZ - 1 - l, t3 = LSZ - 1 - t1;
  const float* oy = out_y + (size_t)b * 4 * 16 * LSZ;
  float y[16];
  float m = 0.f;
#pragma unroll
  for (int c = 0; c < 16; ++c) {
    float dsum = ds[c] + ds[16 + c] + ds[32 + c] + ds[48 + c];
    float v = oy[(0 * 16 + c) * LSZ + t0] + oy[(1 * 16 + c) * LSZ + t1]
            + oy[(2 * 16 + c) * LSZ + t2] + oy[(3 * 16 + c) * LSZ + t3]
            + xconv[(b * 16 + c) * LSZ + l] * dsum;
    y[c] = v; m += v;
  }
  m *= 0.0625f;
  float var = 0.f;
#pragma unroll
  for (int c = 0; c < 16; ++c) { float d = y[c] - m; var += d * d; }
  var *= 0.0625f;
  float rs = rsqrtf(var + 1e-5f);
  float g[16];
#pragma unroll
  for (int c = 0; c < 16; ++c) {
    float yn = (y[c] - m) * rs * onw[c] + onb[c];
    g[c] = yn * siluf(z[idx * 16 + c]);
  }
#pragma unroll
  for (int o = 0; o < 8; ++o) {
    float s = 0.f;
#pragma unroll
    for (int c = 0; c < 16; ++c) s += g[c] * opw[o * 16 + c];
    featOut[idx * 8 + o] = featIn[idx * 8 + o] + s;
  }
}

// ---- kernel 8: emit f (transpose BHWC->BCHW) and f2 (1x1 conv 8->3) into d_out ----
__global__ void k_out(const float* __restrict__ featB, const float* __restrict__ featC,
                      const float* __restrict__ w_out, const float* __restrict__ b_out,
                      float* __restrict__ out) {
  int idx = blockIdx.x * blockDim.x + threadIdx.x;
  if (idx >= BB * LSZ) return;
  int b = idx / LSZ, l = idx - b * LSZ;
  const float* f1 = featB + idx * 8;
#pragma unroll
  for (int o = 0; o < 8; ++o) out[(size_t)(b * 8 + o) * LSZ + l] = f1[o];
  const float* f2 = featC + idx * 8;
  float* o2 = out + (size_t)BB * 8 * LSZ;
#pragma unroll
  for (int o = 0; o < 3; ++o) {
    float s = b_out[o];
#pragma unroll
    for (int c = 0; c < 8; ++c) s += f2[c] * w_out[o * 8 + c];
    o2[(size_t)(b * 3 + o) * LSZ + l] = s;
  }
}

extern "C" void kernel_launch(void* const* d_in, const int* in_sizes, int n_in,
                              void* d_out, int out_size, void* d_ws, size_t ws_size,
                              hipStream_t stream) {
  (void)in_sizes; (void)n_in; (void)out_size; (void)ws_size;
  const float* img   = (const float*)d_in[0];
  const float* w_in  = (const float*)d_in[1];
  const float* b_in  = (const float*)d_in[2];
  const float* w_out = (const float*)d_in[3];
  const float* b_out = (const float*)d_in[4];

  float* ws = (float*)d_ws;
  size_t off = 0;
  auto alloc = [&](size_t nfloats) { float* p = ws + off; off += nfloats; return p; };
  const size_t FA  = (size_t)BB * LSZ * 8;
  const size_t XC  = (size_t)BB * 16 * LSZ;
  const size_t BCS = (size_t)BB * KK * 32 * LSZ;
  const size_t DSs = (size_t)BB * KK * 16 * LSZ;
  const size_t CHs = (size_t)BB * KK * 256 * NC;
  float* featA = alloc(FA);
  float* featB = alloc(FA);
  float* featC = alloc(FA);
  float* xc    = alloc(XC);
  float* zb    = alloc(XC);
  float* xo    = alloc(XC);
  float* BC    = alloc(BCS);
  float* delta = alloc(DSs);
  float* outy  = alloc(DSs);
  float* chA   = alloc(CHs);
  float* chH   = alloc(CHs);
  float* chH0  = alloc(CHs);

  dim3 blk(256);
  const int pixBlocks = (BB * LSZ + 255) / 256;

  k_in_proj<<<pixBlocks, blk, 0, stream>>>(img, w_in, b_in, featA);

  auto vss = [&](int base, const float* fin, float* fout) {
    const float* lnw  = (const float*)d_in[base + 0];
    const float* lnb  = (const float*)d_in[base + 1];
    const float* inw  = (const float*)d_in[base + 2];
    const float* cw   = (const float*)d_in[base + 3];
    const float* cb   = (const float*)d_in[base + 4];
    const float* xpw  = (const float*)d_in[base + 5];
    const float* dtw  = (const float*)d_in[base + 6];
    const float* dtb  = (const float*)d_in[base + 7];
    const float* alog = (const float*)d_in[base + 8];
    const float* dsp  = (const float*)d_in[base + 9];
    const float* onw  = (const float*)d_in[base + 10];
    const float* onb  = (const float*)d_in[base + 11];
    const float* opw  = (const float*)d_in[base + 12];

    k_ln_inproj<<<pixBlocks, blk, 0, stream>>>(fin, lnw, lnb, inw, xc, zb);
    k_dwconv<<<(BB * 16 * LSZ + 255) / 256, blk, 0, stream>>>(xc, cw, cb, xo);
    k_xdbl<<<BB * KK * SEG, blk, 0, stream>>>(xo, xpw, dtw, dtb, BC, delta);
    k_scan1<<<BB * KK * NC, blk, 0, stream>>>(delta, xo, BC, alog, chA, chH);
    k_scan2<<<BB * KK, blk, 0, stream>>>(chA, chH, chH0);
    k_scan3<<<BB * KK * NC, blk, 0, stream>>>(delta, xo, BC, alog, chH0, outy);
    k_combine<<<pixBlocks, blk, 0, stream>>>(outy, xo, zb, dsp, onw, onb, opw, fin, fout);
  };

  vss(5, featA, featB);    // VSS block 1 -> f
  vss(18, featB, featC);   // VSS block 2
  k_out<<<pixBlocks, blk, 0, stream>>>(featB, featC, w_out, b_out, (float*)d_out);
}